// MaxPool_73177652789352
// MI455X (gfx1250) — compile-verified
//
#include <hip/hip_runtime.h>
#include <hip/hip_bf16.h>

// Problem constants (from the reference): B=8, C=256, N=16384, M=4096, K=32
#define B_ 8
#define C_ 256
#define N_ 16384
#define M_ 4096
#define K_ 32
#define CPB_ 2  // channels per workgroup (rows share one index read)

typedef __attribute__((ext_vector_type(4))) unsigned int v4u;
typedef __attribute__((ext_vector_type(8))) int         v8i;
typedef __attribute__((ext_vector_type(4))) int         v4i;

// One workgroup per (b, channel-pair).  The two contiguous 64KB input rows
// input[b,c0,:] and input[b,c0+1,:] are DMA'd into LDS with a single 128KB
// TDM tensor_load_to_lds (issued by wave 0 only — TDM ignores EXEC, so every
// wave would otherwise launch its own DMA).  Each index row indices[b,m,:]
// is then read ONCE and used to gather+max-reduce BOTH channels from LDS,
// halving L2 index traffic versus one-channel-per-workgroup.
__global__ __launch_bounds__(256) void gather_max_kernel(
    const float* __restrict__ input,        // [B, C, N]
    const long long* __restrict__ indices,  // [B, M, K], values in [0, N)
    float* __restrict__ features) {         // [B, C, M]
  __shared__ float rows[CPB_ * N_];  // 128 KB (two contiguous channel rows)

  const int b  = (int)(blockIdx.x >> 7);               // / (C_/CPB_)
  const int c0 = (int)(blockIdx.x & (C_ / CPB_ - 1)) * CPB_;
  const float* src = input + ((size_t)b * C_ + c0) * (size_t)N_;

  // ---- TDM: DMA both 64KB rows (contiguous, 32768 f32) into LDS ----
  if (threadIdx.x < 32) {
    unsigned long long ga = (unsigned long long)(uintptr_t)src;
    // Generic (flat) LDS pointer: low 32 bits are the LDS byte offset.
    unsigned int lds_addr = (unsigned int)(uintptr_t)(&rows[0]);

    // D# group 0: [1:0]=count=1, [63:32]=lds_addr, [120:64]=global_addr,
    //             [127:126]=type=2
    v4u g0 = {1u,
              lds_addr,
              (unsigned int)(ga & 0xffffffffu),
              (unsigned int)((ga >> 32) & 0x01ffffffu) | 0x80000000u};

    // D# group 1 (1-D tile of 32768 4-byte elements):
    //  word0: workgroup_mask=0, data_size=2 (4 bytes)           -> 0x00020000
    //  word1: atomic_barrier_addr=0, tensor_dim0[15:0]=0x8000   -> 0x80000000
    //  word2: tensor_dim0[31:16]=0, tensor_dim1[15:0]=0         -> 0
    //  word3: tensor_dim1[31:16]=0, tile_dim0=0x8000 (32768)    -> 0x80000000
    //  word4: tile_dim1=0 (unused), tile_dim2=0                 -> 0
    //  word5: tensor_dim0_stride[31:0]=32768                    -> 0x00008000
    //  word6,7: stride hi / dim1 stride = 0
    v8i g1 = {(int)0x00020000, (int)0x80000000, 0, (int)0x80000000,
              0,               (int)(CPB_ * N_), 0, 0};

    v4i gz4 = {0, 0, 0, 0};              // groups 2/3: unused (<=2-D tensor)
    v8i gz8 = {0, 0, 0, 0, 0, 0, 0, 0};  // extra group (6-arg toolchain form)

    __builtin_amdgcn_tensor_load_to_lds(g0, g1, gz4, gz4, gz8, /*cpol=*/0);
    __builtin_amdgcn_s_wait_tensorcnt(0);
  }
  __syncthreads();  // other waves wait for wave 0's DMA-complete

  // ---- Gather + max over K neighbors, both channels per index read ----
  // Indices are int64 but values < 16384: read pairs as int4 (b128 loads,
  // 16B aligned since each m-row is 256B) and use the low words (.x, .z).
  const int4* idxb = (const int4*)(indices + (size_t)b * M_ * K_);
  float* out0 = features + ((size_t)b * C_ + c0) * (size_t)M_;
  float* out1 = out0 + M_;

  for (int m = (int)threadIdx.x; m < M_; m += 256) {
    const int4* ip = idxb + (size_t)m * (K_ / 2);
    float best0 = -__builtin_inff();
    float best1 = -__builtin_inff();
#pragma unroll
    for (int k4 = 0; k4 < K_ / 2; ++k4) {
      int4 v = ip[k4];  // two int64 indices (lo words x, z)
      best0 = fmaxf(best0, rows[v.x]);
      best0 = fmaxf(best0, rows[v.z]);
      best1 = fmaxf(best1, rows[N_ + v.x]);
      best1 = fmaxf(best1, rows[N_ + v.z]);
    }
    out0[m] = best0;  // lanes write consecutive m -> fully coalesced
    out1[m] = best1;
  }
}

extern "C" void kernel_launch(void* const* d_in, const int* in_sizes, int n_in,
                              void* d_out, int out_size, void* d_ws, size_t ws_size,
                              hipStream_t stream) {
  const float*     input   = (const float*)d_in[0];      // [B, C, N] f32
  /* d_in[1] = points [B,3,N], unused by the outputs */
  const float*     support = (const float*)d_in[2];      // [B, 3, M] f32
  const long long* indices = (const long long*)d_in[3];  // [B, M, K] i64

  float* out = (float*)d_out;
  float* features    = out;                                        // B*C*M f32
  float* out_support = out + (size_t)B_ * C_ * M_;                 // B*3*M f32
  void*  out_indices = (void*)(out_support + (size_t)B_ * 3 * M_); // B*M*K i64

  // Pass-through outputs: device-to-device async copies (graph-capture safe).
  (void)hipMemcpyAsync(out_support, support, sizeof(float) * (size_t)B_ * 3 * M_,
                       hipMemcpyDeviceToDevice, stream);
  (void)hipMemcpyAsync(out_indices, indices, sizeof(long long) * (size_t)B_ * M_ * K_,
                       hipMemcpyDeviceToDevice, stream);

  // One workgroup per (b, channel-pair): 1024 workgroups x 256 threads,
  // 128KB LDS each (2 workgroups per 320KB WGP -> 16 waves for latency).
  gather_max_kernel<<<dim3(B_ * C_ / CPB_), dim3(256), 0, stream>>>(
      input, indices, features);
}